// PredictionHead_38482906972438
// MI455X (gfx1250) — compile-verified
//
#include <hip/hip_runtime.h>
#include <math.h>

typedef __attribute__((ext_vector_type(2))) float v2f;
typedef __attribute__((ext_vector_type(8))) float v8f;

#define B_DIM    128
#define H_DIM    512
#define MID_DIM  2048
#define NANS     3129
#define N_NODES  131072
#define N_EDGES  262144

#define KCHUNK     64
#define LDS_STRIDE 68   // 64 + pad -> bank-conflict-free LDS reads of A rows

// ---------------------------------------------------------------------------
// Segment-sum pooling over sorted segment ids.
// Block = 256 threads, each block owns `rows_per_block` contiguous rows.
// Each thread accumulates 2 columns in registers; atomicAdd only when the
// segment id changes (segments are sorted, so flushes are rare).
// ---------------------------------------------------------------------------
__global__ __launch_bounds__(256)
void segpool_kernel(const float* __restrict__ feat, const int* __restrict__ seg,
                    float* __restrict__ sum, float* __restrict__ cnt,
                    int n_rows, int rows_per_block)
{
    const int r0  = blockIdx.x * rows_per_block;
    int r1        = r0 + rows_per_block;
    if (r1 > n_rows) r1 = n_rows;
    if (r0 >= r1) return;

    const int tid = threadIdx.x;     // 0..255
    const int c   = tid * 2;         // two columns per thread (H=512)

    float acc0 = 0.0f, acc1 = 0.0f, count = 0.0f;
    int cur = seg[r0];

    for (int r = r0; r < r1; ++r) {
        const int s = seg[r];
        if (s != cur) {
            atomicAdd(&sum[cur * H_DIM + c],     acc0);
            atomicAdd(&sum[cur * H_DIM + c + 1], acc1);
            if (tid == 0) atomicAdd(&cnt[cur], count);
            acc0 = 0.0f; acc1 = 0.0f; count = 0.0f;
            cur = s;
        }
        const float2* row = (const float2*)(feat + (size_t)r * H_DIM);
        const float2 v = row[tid];
        acc0 += v.x;
        acc1 += v.y;
        count += 1.0f;
        if (r + 8 < n_rows) {
            __builtin_prefetch(feat + (size_t)(r + 8) * H_DIM + c, 0, 1);
        }
    }
    atomicAdd(&sum[cur * H_DIM + c],     acc0);
    atomicAdd(&sum[cur * H_DIM + c + 1], acc1);
    if (tid == 0) atomicAdd(&cnt[cur], count);
}

// ---------------------------------------------------------------------------
// fp32 WMMA 16x16 tile mainloop: one wave owns one 16x16 output tile.
// A panel (16 x KCHUNK) staged in LDS cooperatively by the whole block.
// f32 WMMA operand layout (CDNA5 ISA 7.12.2):
//   A 16x4: lanes 0-15 -> M=lane, v0=K0,v1=K1 ; lanes 16-31 -> M=lane-16, K2/K3
//   B 4x16: lanes 0-15 -> N=lane, v0=K0,v1=K1 ; lanes 16-31 -> N=lane-16, K2/K3
//   C/D:    reg r, lanes 0-15 -> M=r, N=lane ; lanes 16-31 -> M=8+r, N=lane-16
// ---------------------------------------------------------------------------
template<bool BOUNDED>
__device__ inline v8f wmma_tile_f32(const float* __restrict__ A, int K,
                                    const float* __restrict__ Bw, int ldb,
                                    int ncolbase, int nlimit,
                                    int m0, float* As)
{
    const int lane = threadIdx.x & 31;
    const int half = lane >> 4;
    const int l16  = lane & 15;
    const int col  = ncolbase + l16;

    v8f acc = {};
    for (int kc = 0; kc < K; kc += KCHUNK) {
        // cooperative stage: 16 rows x 64 k-values = 1024 floats / 256 threads
        {
            const int e  = threadIdx.x * 4;
            const int m  = e >> 6;       // / KCHUNK
            const int kk = e & 63;       // % KCHUNK
            const float4 av = *(const float4*)(A + (size_t)(m0 + m) * K + kc + kk);
            float* dst = &As[m * LDS_STRIDE + kk];
            dst[0] = av.x; dst[1] = av.y; dst[2] = av.z; dst[3] = av.w;
        }
        __syncthreads();

        const float* arow = &As[l16 * LDS_STRIDE];
        for (int kk4 = 0; kk4 < KCHUNK; kk4 += 4) {
            v2f a, b;
            a.x = arow[kk4 + half * 2];
            a.y = arow[kk4 + half * 2 + 1];
            const int kg = kc + kk4 + half * 2;
            if (BOUNDED) {
                b.x = (col < nlimit) ? Bw[(size_t)kg * ldb + col]       : 0.0f;
                b.y = (col < nlimit) ? Bw[(size_t)(kg + 1) * ldb + col] : 0.0f;
            } else {
                b.x = Bw[(size_t)kg * ldb + col];
                b.y = Bw[(size_t)(kg + 1) * ldb + col];
            }
            acc = __builtin_amdgcn_wmma_f32_16x16x4_f32(
                false, a, false, b, (short)0, acc, false, false);
        }
        __syncthreads();
    }
    return acc;
}

// ---------------------------------------------------------------------------
// GEMM1 (fused): h_pre[128,1024]
//   cols [0,512):   2*(q @ Wqe + bqe) + 2*mean_img_edge
//   cols [512,1024):2*(q @ Wqn + bqn) + mean_img_node + mean_kg_node
// ---------------------------------------------------------------------------
__global__ __launch_bounds__(256)
void k_gemm_fuse(const float* __restrict__ q,
                 const float* __restrict__ Wqe, const float* __restrict__ bqe,
                 const float* __restrict__ Wqn, const float* __restrict__ bqn,
                 const float* __restrict__ sum_ie, const float* __restrict__ cnt_ie,
                 const float* __restrict__ sum_in, const float* __restrict__ cnt_in,
                 const float* __restrict__ sum_kn, const float* __restrict__ cnt_kn,
                 float* __restrict__ hpre)
{
    __shared__ float As[16 * LDS_STRIDE];
    const int wave = threadIdx.x >> 5;
    const int lane = threadIdx.x & 31;
    const int half = lane >> 4;
    const int l16  = lane & 15;
    const int m0   = blockIdx.y * 16;
    const int n0   = (blockIdx.x * 8 + wave) * 16;

    const bool edge = (n0 < H_DIM);                  // tile never straddles 512
    const float* Wm = edge ? Wqe : Wqn;
    const int ncol  = edge ? n0 : (n0 - H_DIM);

    v8f acc = wmma_tile_f32<false>(q, 2 * H_DIM, Wm, H_DIM, ncol, 0, m0, As);

    const int c = n0 + l16;
    for (int r = 0; r < 8; ++r) {
        const int m = m0 + r + half * 8;             // batch index
        const float v = acc[r];
        float o;
        if (edge) {
            const float mean = sum_ie[m * H_DIM + c] / fmaxf(cnt_ie[m], 1.0f);
            o = 2.0f * (v + bqe[c]) + 2.0f * mean;
        } else {
            const int nn = c - H_DIM;
            const float mi = sum_in[m * H_DIM + nn] / fmaxf(cnt_in[m], 1.0f);
            const float mk = sum_kn[m * H_DIM + nn] / fmaxf(cnt_kn[m], 1.0f);
            o = 2.0f * (v + bqn[nn]) + mi + mk;
        }
        hpre[(size_t)m * (2 * H_DIM) + c] = o;
    }
}

// ---------------------------------------------------------------------------
// GEMM2: h_mid = elu(h_pre @ W1 + b1)   [128,2048]
// ---------------------------------------------------------------------------
__global__ __launch_bounds__(256)
void k_gemm_elu(const float* __restrict__ A,
                const float* __restrict__ W1, const float* __restrict__ b1,
                float* __restrict__ hmid)
{
    __shared__ float As[16 * LDS_STRIDE];
    const int wave = threadIdx.x >> 5;
    const int lane = threadIdx.x & 31;
    const int half = lane >> 4;
    const int l16  = lane & 15;
    const int m0   = blockIdx.y * 16;
    const int n0   = (blockIdx.x * 8 + wave) * 16;

    v8f acc = wmma_tile_f32<false>(A, 2 * H_DIM, W1, MID_DIM, n0, 0, m0, As);

    const int c = n0 + l16;
    for (int r = 0; r < 8; ++r) {
        const int m = m0 + r + half * 8;
        const float x = acc[r] + b1[c];
        hmid[(size_t)m * MID_DIM + c] = (x > 0.0f) ? x : expm1f(x);
    }
}

// ---------------------------------------------------------------------------
// GEMM3: out = h_mid @ W2 + b2   [128,3129]  (N not multiple of 16 -> bounded)
// ---------------------------------------------------------------------------
__global__ __launch_bounds__(256)
void k_gemm_out(const float* __restrict__ A,
                const float* __restrict__ W2, const float* __restrict__ b2,
                float* __restrict__ out)
{
    __shared__ float As[16 * LDS_STRIDE];
    const int wave = threadIdx.x >> 5;
    const int lane = threadIdx.x & 31;
    const int half = lane >> 4;
    const int l16  = lane & 15;
    const int m0   = blockIdx.y * 16;
    const int n0   = (blockIdx.x * 8 + wave) * 16;

    v8f acc = wmma_tile_f32<true>(A, MID_DIM, W2, NANS, n0, NANS, m0, As);

    const int c = n0 + l16;
    if (c < NANS) {
        for (int r = 0; r < 8; ++r) {
            const int m = m0 + r + half * 8;
            out[(size_t)m * NANS + c] = acc[r] + b2[c];
        }
    }
}

// ---------------------------------------------------------------------------
// Workspace layout (floats):
//   [0]        sum_img_node  128*512
//   [65536]    sum_kg_node   128*512
//   [131072]   sum_img_edge  128*512
//   [196608]   cnt_img_node  128
//   [196736]   cnt_kg_node   128
//   [196864]   cnt_img_edge  128
//   [196992]   h_pre         128*1024
//   [328064]   h_mid         128*2048
// ---------------------------------------------------------------------------
extern "C" void kernel_launch(void* const* d_in, const int* in_sizes, int n_in,
                              void* d_out, int out_size, void* d_ws, size_t ws_size,
                              hipStream_t stream)
{
    const float* question      = (const float*)d_in[0];
    const float* img_node_feat = (const float*)d_in[1];
    const float* kg_node_feat  = (const float*)d_in[2];
    const float* img_edge_feat = (const float*)d_in[3];
    // d_in[4] kg_edge_feat: unused (reference bug pools kg_edges from img edges)
    const int*   img_node_seg  = (const int*)d_in[5];
    const int*   kg_node_seg   = (const int*)d_in[6];
    const int*   img_edge_seg  = (const int*)d_in[7];
    // d_in[8] kg_edge_seg: unused
    const float* Wqe = (const float*)d_in[9];
    const float* bqe = (const float*)d_in[10];
    const float* Wqn = (const float*)d_in[11];
    const float* bqn = (const float*)d_in[12];
    const float* W1  = (const float*)d_in[13];
    const float* b1  = (const float*)d_in[14];
    const float* W2  = (const float*)d_in[15];
    const float* b2  = (const float*)d_in[16];

    float* ws       = (float*)d_ws;
    float* sum_in_  = ws;
    float* sum_kn_  = ws + 65536;
    float* sum_ie_  = ws + 131072;
    float* cnt_in_  = ws + 196608;
    float* cnt_kn_  = ws + 196736;
    float* cnt_ie_  = ws + 196864;
    float* h_pre    = ws + 196992;
    float* h_mid    = ws + 328064;

    // zero the accumulation region (sums + counts) each call: deterministic
    hipMemsetAsync(ws, 0, (size_t)196992 * sizeof(float), stream);

    // ---- pooling (bandwidth-bound: ~1 GB streamed) ----
    const int RPB = 256;
    segpool_kernel<<<N_NODES / RPB, 256, 0, stream>>>(img_node_feat, img_node_seg,
                                                      sum_in_, cnt_in_, N_NODES, RPB);
    segpool_kernel<<<N_NODES / RPB, 256, 0, stream>>>(kg_node_feat, kg_node_seg,
                                                      sum_kn_, cnt_kn_, N_NODES, RPB);
    segpool_kernel<<<N_EDGES / RPB, 256, 0, stream>>>(img_edge_feat, img_edge_seg,
                                                      sum_ie_, cnt_ie_, N_EDGES, RPB);

    // ---- fused head (fp32 WMMA) ----
    k_gemm_fuse<<<dim3((2 * H_DIM) / 128, B_DIM / 16), 256, 0, stream>>>(
        question, Wqe, bqe, Wqn, bqn,
        sum_ie_, cnt_ie_, sum_in_, cnt_in_, sum_kn_, cnt_kn_, h_pre);

    k_gemm_elu<<<dim3(MID_DIM / 128, B_DIM / 16), 256, 0, stream>>>(
        h_pre, W1, b1, h_mid);

    const int n_tiles3 = (NANS + 15) / 16;               // 196
    k_gemm_out<<<dim3((n_tiles3 + 7) / 8, B_DIM / 16), 256, 0, stream>>>(
        h_mid, W2, b2, (float*)d_out);
}